// UniGCNConv_50749333569736
// MI455X (gfx1250) — compile-verified
//
#include <hip/hip_runtime.h>
#include <hip/hip_bf16.h>

typedef __attribute__((ext_vector_type(2))) float v2f;
typedef __attribute__((ext_vector_type(8))) float v8f;

#define C_DIM 256        // C_IN == C_OUT == 256
#define TILE  16
#define NT    4                          // N-tiles per wave (16x64 strip)
#define NGROUPS (C_DIM / (TILE * NT))    // 4 column groups

// ---------------------------------------------------------------------------
// Zero-fill (float4 stores; n4 = element-count/4; all our sizes divide by 4)
// ---------------------------------------------------------------------------
__global__ void zero_f32(float* __restrict__ p, long n4) {
    long i = (long)blockIdx.x * blockDim.x + threadIdx.x;
    if (i < n4) {
        float4 z = make_float4(0.f, 0.f, 0.f, 0.f);
        ((float4*)p)[i] = z;
    }
}

// ---------------------------------------------------------------------------
// Xw = X @ W.T  via V_WMMA_F32_16X16X4_F32 (exact fp32, matches reference)
// One wave computes a 16x64 strip: 4 accumulators share one A fragment per
// k-step (A reuse in registers -> 4x less X traffic, 4 independent WMMAs per
// A load for latency hiding).
// A-frag (32-bit 16x4): lanes 0-15 -> M=lane, K={0,1}; lanes 16-31 -> K={2,3}.
// B-frag (4x16): B[k][n] = W[n][k]; W row-major, so each lane's K-pair is a
// contiguous b64. The 4 B tiles sit at constant 16KB strides (inst offsets).
// C/D 16x16 f32: VGPR g holds row (g + 8*(lane>>4)), col lane&15.
// ---------------------------------------------------------------------------
__global__ __launch_bounds__(128) void gemm_xw_wmma(
        const float* __restrict__ X, const float* __restrict__ W,
        float* __restrict__ Xw, int M) {
    const int lane = threadIdx.x & 31;
    const int wave = threadIdx.x >> 5;
    const int gid  = blockIdx.x * 4 + wave;      // one 16x64 strip per wave
    const int m0 = (gid / NGROUPS) * TILE;
    const int n0 = (gid % NGROUPS) * (TILE * NT);
    if (m0 >= M) return;

    const int r    = lane & 15;
    const int half = lane >> 4;

    const float* Arow = X + (size_t)(m0 + r) * C_DIM + half * 2;
    const float* Brow = W + (size_t)(n0 + r) * C_DIM + half * 2;

    v8f acc[NT] = {};
#pragma unroll 4
    for (int kk = 0; kk < C_DIM / 4; ++kk) {
        v2f a = *(const v2f*)(Arow + kk * 4);
        v2f b[NT];
#pragma unroll
        for (int t = 0; t < NT; ++t)
            b[t] = *(const v2f*)(Brow + (size_t)t * TILE * C_DIM + kk * 4);
#pragma unroll
        for (int t = 0; t < NT; ++t)
            // 8 args: (neg_a, A, neg_b, B, c_mod, C, reuse_a, reuse_b)
            acc[t] = __builtin_amdgcn_wmma_f32_16x16x4_f32(
                    false, a, false, b[t], (short)0, acc[t], false, false);
    }

    float* Crow = Xw + (size_t)(m0 + half * 8) * C_DIM + n0 + r;
#pragma unroll
    for (int t = 0; t < NT; ++t)
#pragma unroll
        for (int g = 0; g < 8; ++g)
            Crow[(size_t)g * C_DIM + t * TILE] = acc[t][g];
}

// ---------------------------------------------------------------------------
// Edge aggregation: one wave per incidence entry.
// Gather row Xw[vertex[i]] (L2-resident, 51 MB << 192 MB L2) and atomically
// accumulate into Xe_sum[edges[i]]; lane 0 bumps the edge count.
// ---------------------------------------------------------------------------
__global__ __launch_bounds__(256) void scatter_edges(
        const float* __restrict__ Xw, const int* __restrict__ vertex,
        const int* __restrict__ edges, float* __restrict__ Xe_sum,
        float* __restrict__ cnt, int nnz) {
    const int wid  = (int)(((long)blockIdx.x * blockDim.x + threadIdx.x) >> 5);
    const int lane = threadIdx.x & 31;
    if (wid >= nnz) return;
    const int v = vertex[wid];
    const int e = edges[wid];
    const float4* src = (const float4*)(Xw + (size_t)v * C_DIM);
    float* dst = Xe_sum + (size_t)e * C_DIM;
#pragma unroll
    for (int j = 0; j < 2; ++j) {
        const int q = lane + j * 32;          // 64 float4 per row
        float4 x = src[q];
        const int c = q * 4;
        atomicAdd(dst + c + 0, x.x);
        atomicAdd(dst + c + 1, x.y);
        atomicAdd(dst + c + 2, x.z);
        atomicAdd(dst + c + 3, x.w);
    }
    if (lane == 0) atomicAdd(cnt + e, 1.0f);
}

// ---------------------------------------------------------------------------
// Per-edge composite scale: s[e] = dege[e] / max(cnt[e], 1)
// ---------------------------------------------------------------------------
__global__ void edge_scale(const float* __restrict__ cnt,
                           const float* __restrict__ dege,
                           float* __restrict__ s, int E) {
    int e = blockIdx.x * blockDim.x + threadIdx.x;
    if (e < E) s[e] = dege[e] / fmaxf(cnt[e], 1.0f);
}

// ---------------------------------------------------------------------------
// Vertex aggregation: one wave per incidence entry.
// val = Xe_sum[e] * s[e]  (fuses mean-divide + dege scale), atomic add into
// out[vertex[i]].
// ---------------------------------------------------------------------------
__global__ __launch_bounds__(256) void scatter_vertices(
        const float* __restrict__ Xe_sum, const float* __restrict__ s,
        const int* __restrict__ vertex, const int* __restrict__ edges,
        float* __restrict__ out, int nnz) {
    const int wid  = (int)(((long)blockIdx.x * blockDim.x + threadIdx.x) >> 5);
    const int lane = threadIdx.x & 31;
    if (wid >= nnz) return;
    const int v = vertex[wid];
    const int e = edges[wid];
    const float se = s[e];
    const float4* src = (const float4*)(Xe_sum + (size_t)e * C_DIM);
    float* dst = out + (size_t)v * C_DIM;
#pragma unroll
    for (int j = 0; j < 2; ++j) {
        const int q = lane + j * 32;
        float4 x = src[q];
        const int c = q * 4;
        atomicAdd(dst + c + 0, x.x * se);
        atomicAdd(dst + c + 1, x.y * se);
        atomicAdd(dst + c + 2, x.z * se);
        atomicAdd(dst + c + 3, x.w * se);
    }
}

// ---------------------------------------------------------------------------
// Final per-vertex degree scale: out[v][c] *= degv[v]
// ---------------------------------------------------------------------------
__global__ void scale_out(float* __restrict__ out,
                          const float* __restrict__ degv, long total) {
    long i = (long)blockIdx.x * blockDim.x + threadIdx.x;
    if (i < total) out[i] *= degv[i >> 8];   // C_DIM == 256
}

// ---------------------------------------------------------------------------
extern "C" void kernel_launch(void* const* d_in, const int* in_sizes, int n_in,
                              void* d_out, int out_size, void* d_ws, size_t ws_size,
                              hipStream_t stream) {
    const float* X      = (const float*)d_in[0];
    const int*   vertex = (const int*)  d_in[1];
    const int*   edges  = (const int*)  d_in[2];
    const float* W      = (const float*)d_in[3];
    const float* dege   = (const float*)d_in[4];
    const float* degv   = (const float*)d_in[5];
    float*       out    = (float*)d_out;

    const int N   = in_sizes[0] / C_DIM;   // 50000 vertices
    const int nnz = in_sizes[1];           // 800000 incidence entries
    const int E   = in_sizes[4];           // 50000 hyperedges

    // Workspace layout
    char* ws = (char*)d_ws;
    float* Xw     = (float*)ws;                                   // N * 256 f32
    float* Xe_sum = (float*)(ws + (size_t)N * C_DIM * 4);         // E * 256 f32
    float* cnt    = (float*)(ws + (size_t)(N + E) * C_DIM * 4);   // E f32
    float* s      = cnt + E;                                      // E f32

    // 1) zero accumulators + output
    {
        long n4;
        n4 = (long)E * C_DIM / 4;
        hipLaunchKernelGGL(zero_f32, dim3((n4 + 255) / 256), dim3(256), 0, stream, Xe_sum, n4);
        n4 = (long)E / 4;
        hipLaunchKernelGGL(zero_f32, dim3((n4 + 255) / 256), dim3(256), 0, stream, cnt, n4);
        n4 = (long)N * C_DIM / 4;
        hipLaunchKernelGGL(zero_f32, dim3((n4 + 255) / 256), dim3(256), 0, stream, out, n4);
    }

    // 2) Xw = X @ W.T  (fp32 WMMA, 16x64 strip per wave)
    {
        const int strips = (N / TILE) * NGROUPS;          // 3125 * 4 = 12500
        const int blocks = (strips + 3) / 4;              // 4 waves / block
        hipLaunchKernelGGL(gemm_xw_wmma, dim3(blocks), dim3(128), 0, stream, X, W, Xw, N);
    }

    // 3) gather-by-vertex, atomic sum into edges (+ counts)
    {
        const int blocks = (int)(((long)nnz * 32 + 255) / 256);
        hipLaunchKernelGGL(scatter_edges, dim3(blocks), dim3(256), 0, stream,
                           Xw, vertex, edges, Xe_sum, cnt, nnz);
    }

    // 4) per-edge scale factor
    hipLaunchKernelGGL(edge_scale, dim3((E + 255) / 256), dim3(256), 0, stream,
                       cnt, dege, s, E);

    // 5) gather-by-edge (scaled), atomic sum into vertices
    {
        const int blocks = (int)(((long)nnz * 32 + 255) / 256);
        hipLaunchKernelGGL(scatter_vertices, dim3(blocks), dim3(256), 0, stream,
                           Xe_sum, s, vertex, edges, out, nnz);
    }

    // 6) final degv scale
    {
        const long total = (long)N * C_DIM;
        hipLaunchKernelGGL(scale_out, dim3((total + 255) / 256), dim3(256), 0, stream,
                           out, degv, total);
    }
}